// HeteroGAT_44779329028271
// MI455X (gfx1250) — compile-verified
//
#include <hip/hip_runtime.h>

typedef float v2f __attribute__((ext_vector_type(2)));
typedef float v8f __attribute__((ext_vector_type(8)));

#define HEADS 4
#define DHEAD 32
#define FDIM  128
#define NEG_SLOPE 0.2f

// ---------------------------------------------------------------------------
// WMMA GEMM: C[N,128] = A[N,128] @ W[128,128], fp32 via V_WMMA_F32_16X16X4_F32
// Block = 256 threads = 8 waves; wave w -> 16-col tile, block -> 16-row tile.
// Out-of-range rows are CLAMPED (not predicated) so EXEC stays full and the
// A/B loads are unconditional -> clause-batched + pipelined across unroll-8.
// A frag (16x4 f32): lanes 0-15 row M=lane, lanes 16-31 row M=lane-16;
//   VGPR0 = K=(0|2)+k0, VGPR1 = K=(1|3)+k0 depending on lane-half.
// C frag (16x16 f32): VGPR v: lanes0-15 M=v, lanes16-31 M=v+8, N=lane&15.
// ---------------------------------------------------------------------------
__global__ __launch_bounds__(256)
void gemm128_wmma(const float* __restrict__ A, const float* __restrict__ W,
                  float* __restrict__ C, int N) {
  const int wave = threadIdx.x >> 5;
  const int lane = threadIdx.x & 31;
  const int hl   = lane >> 4;          // lane half (0/1)
  const int r    = lane & 15;
  const int row0 = blockIdx.x << 4;
  const int col0 = wave << 4;
  int aRow = row0 + r;
  aRow = aRow < N ? aRow : N - 1;      // clamp: garbage rows computed, never stored
  const float* Ap = A + (size_t)aRow * FDIM + 2 * hl;
  const float* Wp = W + (size_t)(2 * hl) * FDIM + col0 + r;

  v8f acc = {0.f, 0.f, 0.f, 0.f, 0.f, 0.f, 0.f, 0.f};
#pragma unroll 8
  for (int k0 = 0; k0 < FDIM; k0 += 4) {
    const float2 t = *reinterpret_cast<const float2*>(Ap + k0);
    v2f a; a.x = t.x; a.y = t.y;
    v2f b;
    b.x = Wp[(size_t)k0 * FDIM];
    b.y = Wp[(size_t)k0 * FDIM + FDIM];
    acc = __builtin_amdgcn_wmma_f32_16x16x4_f32(false, a, false, b,
                                                (short)0, acc, false, false);
  }
#pragma unroll
  for (int v = 0; v < 8; ++v) {
    const int row = row0 + v + 8 * hl;
    if (row < N) C[(size_t)row * FDIM + col0 + r] = acc[v];
  }
}

// ve[k,h] = sum_d We[k, h*32+d] * ae[h,d]   (tiny: 16x4)
__global__ void ve_kernel(const float* __restrict__ We,
                          const float* __restrict__ ae,
                          float* __restrict__ ve) {
  const int t = threadIdx.x;
  if (t >= 16 * HEADS) return;
  const int k = t >> 2, h = t & 3;
  float s = 0.f;
  for (int d = 0; d < DHEAD; ++d)
    s += We[k * FDIM + h * DHEAD + d] * ae[h * DHEAD + d];
  ve[k * HEADS + h] = s;
}

// out[n,h] = sum_d P[n, h*32+d] * avec[h,d]
__global__ void node_attn(const float* __restrict__ P,
                          const float* __restrict__ avec,
                          float* __restrict__ out, int N) {
  const int idx = blockIdx.x * blockDim.x + threadIdx.x;
  if (idx >= N * HEADS) return;
  const int n = idx >> 2, h = idx & 3;
  const float4* p = reinterpret_cast<const float4*>(P + (size_t)n * FDIM + h * DHEAD);
  const float4* a = reinterpret_cast<const float4*>(avec + h * DHEAD);
  float s = 0.f;
#pragma unroll
  for (int i = 0; i < 8; ++i) {
    const float4 pv = p[i], av = a[i];
    s += pv.x * av.x + pv.y * av.y + pv.z * av.z + pv.w * av.w;
  }
  out[idx] = s;
}

// monotonic float -> uint key for atomic segment-max
__device__ __forceinline__ unsigned fkey(float x) {
  unsigned u = __float_as_uint(x);
  return (u & 0x80000000u) ? ~u : (u | 0x80000000u);
}

// per-edge: ee = Ef . ve ; score = lrelu(el[src]+er[dst]+ee) ; atomicMax key
__global__ void edge_scores(const float* __restrict__ Ef,
                            const float* __restrict__ ve,
                            const float* __restrict__ el,
                            const float* __restrict__ er,
                            const int* __restrict__ src,
                            const int* __restrict__ dst,
                            float* __restrict__ scores,
                            unsigned* __restrict__ mkey, int E) {
  const int e = blockIdx.x * blockDim.x + threadIdx.x;
  if (e >= E) return;
  const float4* efp = reinterpret_cast<const float4*>(Ef + (size_t)e * 16);
  float ef[16];
#pragma unroll
  for (int i = 0; i < 4; ++i) {
    const float4 v = efp[i];
    ef[4 * i] = v.x; ef[4 * i + 1] = v.y; ef[4 * i + 2] = v.z; ef[4 * i + 3] = v.w;
  }
  const int s = src[e], d = dst[e];
#pragma unroll
  for (int h = 0; h < HEADS; ++h) {
    float ee = 0.f;
#pragma unroll
    for (int k = 0; k < 16; ++k) ee += ef[k] * ve[k * HEADS + h];
    float sc = el[s * HEADS + h] + er[d * HEADS + h] + ee;
    sc = sc > 0.f ? sc : NEG_SLOPE * sc;
    scores[(size_t)e * HEADS + h] = sc;
    atomicMax(&mkey[d * HEADS + h], fkey(sc));
  }
}

// decode key -> float; non-finite (no incoming edge) -> 0   (in place)
__global__ void fix_m(unsigned* __restrict__ mkey, int NH) {
  const int i = blockIdx.x * blockDim.x + threadIdx.x;
  if (i >= NH) return;
  const unsigned u = mkey[i];
  const unsigned bits = (u & 0x80000000u) ? (u & 0x7FFFFFFFu) : ~u;
  float v = __uint_as_float(bits);
  if (!(fabsf(v) < 3.3e38f)) v = 0.f;   // NaN / Inf -> 0
  reinterpret_cast<float*>(mkey)[i] = v;
}

// one wave per edge: lane = feature dim within head (coalesced 128B chunks)
__global__ __launch_bounds__(256)
void edge_scatter(const float* __restrict__ scores, const float* __restrict__ m,
                  const float* __restrict__ fs,
                  const int* __restrict__ src, const int* __restrict__ dst,
                  float* __restrict__ ssum, float* __restrict__ msg, int E) {
  const int wave = threadIdx.x >> 5;
  const int lane = threadIdx.x & 31;
  const int e = blockIdx.x * 8 + wave;
  if (e >= E) return;
  const int si = src[e], di = dst[e];
  float w[HEADS];
#pragma unroll
  for (int h = 0; h < HEADS; ++h)
    w[h] = expf(scores[(size_t)e * HEADS + h] - m[di * HEADS + h]);
  if (lane == 0) {
#pragma unroll
    for (int h = 0; h < HEADS; ++h) atomicAdd(&ssum[di * HEADS + h], w[h]);
  }
  const float* fsp = fs + (size_t)si * FDIM;
  float* mp = msg + (size_t)di * FDIM;
#pragma unroll
  for (int h = 0; h < HEADS; ++h)
    atomicAdd(&mp[h * DHEAD + lane], w[h] * fsp[h * DHEAD + lane]);
}

// out = msg / (s + 1e-9), optional ReLU, in place
__global__ void norm_kernel(float* __restrict__ msg, const float* __restrict__ ssum,
                            int N, int doRelu) {
  const int i = blockIdx.x * blockDim.x + threadIdx.x;
  if (i >= N * FDIM) return;
  const int n = i >> 7, c = i & 127, h = c >> 5;
  float v = msg[i] / (ssum[n * HEADS + h] + 1e-9f);
  if (doRelu) v = v > 0.f ? v : 0.f;
  msg[i] = v;
}

// ---------------------------------------------------------------------------
static void run_edge_gat(const float* Asrc, int Ns, const float* Adst, int Nd,
                         const float* W, const float* We,
                         const float* al, const float* ar, const float* ae,
                         const float* Ef, const int* src, const int* dst, int E,
                         float* outBuf, int doRelu,
                         float* fsb, float* fdb, float* elb, float* erb,
                         float* sb, unsigned* mb, float* scb, float* veb,
                         hipStream_t stream) {
  const dim3 blk(256);
  gemm128_wmma<<<dim3((Ns + 15) / 16), blk, 0, stream>>>(Asrc, W, fsb, Ns);
  gemm128_wmma<<<dim3((Nd + 15) / 16), blk, 0, stream>>>(Adst, W, fdb, Nd);
  ve_kernel<<<1, 64, 0, stream>>>(We, ae, veb);
  node_attn<<<dim3((Ns * HEADS + 255) / 256), blk, 0, stream>>>(fsb, al, elb, Ns);
  node_attn<<<dim3((Nd * HEADS + 255) / 256), blk, 0, stream>>>(fdb, ar, erb, Nd);
  hipMemsetAsync(mb, 0, sizeof(unsigned) * (size_t)Nd * HEADS, stream);
  edge_scores<<<dim3((E + 255) / 256), blk, 0, stream>>>(Ef, veb, elb, erb, src, dst,
                                                         scb, mb, E);
  fix_m<<<dim3((Nd * HEADS + 255) / 256), blk, 0, stream>>>(mb, Nd * HEADS);
  hipMemsetAsync(sb, 0, sizeof(float) * (size_t)Nd * HEADS, stream);
  hipMemsetAsync(outBuf, 0, sizeof(float) * (size_t)Nd * FDIM, stream);
  edge_scatter<<<dim3((E + 7) / 8), blk, 0, stream>>>(scb, (const float*)mb, fsb,
                                                      src, dst, sb, outBuf, E);
  norm_kernel<<<dim3(((unsigned)Nd * FDIM + 255) / 256), blk, 0, stream>>>(
      outBuf, sb, Nd, doRelu);
}

extern "C" void kernel_launch(void* const* d_in, const int* in_sizes, int n_in,
                              void* d_out, int out_size, void* d_ws, size_t ws_size,
                              hipStream_t stream) {
  const float* Xu = (const float*)d_in[0];
  const float* Xi = (const float*)d_in[1];
  const float* Ef_ui0 = (const float*)d_in[2];
  const float* Ef_iu0 = (const float*)d_in[3];
  const float* Ef_ui1 = (const float*)d_in[4];
  const float* Ef_iu1 = (const float*)d_in[5];
  // parameter blocks: [W, We, al, ar, ae] for ui0, iu0, ui1, iu1
  const float *W_ui0 = (const float*)d_in[6],  *We_ui0 = (const float*)d_in[7],
              *al_ui0 = (const float*)d_in[8], *ar_ui0 = (const float*)d_in[9],
              *ae_ui0 = (const float*)d_in[10];
  const float *W_iu0 = (const float*)d_in[11], *We_iu0 = (const float*)d_in[12],
              *al_iu0 = (const float*)d_in[13], *ar_iu0 = (const float*)d_in[14],
              *ae_iu0 = (const float*)d_in[15];
  const float *W_ui1 = (const float*)d_in[16], *We_ui1 = (const float*)d_in[17],
              *al_ui1 = (const float*)d_in[18], *ar_ui1 = (const float*)d_in[19],
              *ae_ui1 = (const float*)d_in[20];
  const float *W_iu1 = (const float*)d_in[21], *We_iu1 = (const float*)d_in[22],
              *al_iu1 = (const float*)d_in[23], *ar_iu1 = (const float*)d_in[24],
              *ae_iu1 = (const float*)d_in[25];
  const int* src_ui = (const int*)d_in[26];
  const int* dst_ui = (const int*)d_in[27];
  const int* src_iu = (const int*)d_in[28];
  const int* dst_iu = (const int*)d_in[29];

  const int NU = in_sizes[0] / FDIM;
  const int NI = in_sizes[1] / FDIM;
  const int E  = in_sizes[26];
  const int Nmax = NU > NI ? NU : NI;

  float* ws = (float*)d_ws;
  size_t off = 0;
  float* fsb = ws + off; off += (size_t)Nmax * FDIM;
  float* fdb = ws + off; off += (size_t)Nmax * FDIM;
  float* hu1 = ws + off; off += (size_t)NU * FDIM;
  float* hi1 = ws + off; off += (size_t)NI * FDIM;
  float* elb = ws + off; off += (size_t)Nmax * HEADS;
  float* erb = ws + off; off += (size_t)Nmax * HEADS;
  float* sb  = ws + off; off += (size_t)Nmax * HEADS;
  unsigned* mb = (unsigned*)(ws + off); off += (size_t)Nmax * HEADS;
  float* scb = ws + off; off += (size_t)E * HEADS;
  float* veb = ws + off; off += 64;

  float* hu2 = (float*)d_out;                      // outputs: (hu2, hi2)
  float* hi2 = (float*)d_out + (size_t)NU * FDIM;

  // ---- layer 0 (ReLU on outputs) ----
  run_edge_gat(Xu, NU, Xi, NI, W_ui0, We_ui0, al_ui0, ar_ui0, ae_ui0,
               Ef_ui0, src_ui, dst_ui, E, hi1, 1,
               fsb, fdb, elb, erb, sb, mb, scb, veb, stream);
  run_edge_gat(Xi, NI, Xu, NU, W_iu0, We_iu0, al_iu0, ar_iu0, ae_iu0,
               Ef_iu0, src_iu, dst_iu, E, hu1, 1,
               fsb, fdb, elb, erb, sb, mb, scb, veb, stream);
  // ---- layer 1 (no ReLU, write straight into d_out) ----
  run_edge_gat(hu1, NU, hi1, NI, W_ui1, We_ui1, al_ui1, ar_ui1, ae_ui1,
               Ef_ui1, src_ui, dst_ui, E, hi2, 0,
               fsb, fdb, elb, erb, sb, mb, scb, veb, stream);
  run_edge_gat(hi1, NI, hu1, NU, W_iu1, We_iu1, al_iu1, ar_iu1, ae_iu1,
               Ef_iu1, src_iu, dst_iu, E, hu2, 0,
               fsb, fdb, elb, erb, sb, mb, scb, veb, stream);
}